// Quantizer_88940182766354
// MI455X (gfx1250) — compile-verified
//
#include <hip/hip_runtime.h>

// VQ-VAE quantizer for MI455X (gfx1250, wave32).
// Heavy op: 65536x2048x256 fp32 GEMM -> row argmin, done on V_WMMA_F32_16X16X4_F32.

typedef __attribute__((ext_vector_type(2))) float v2f;
typedef __attribute__((ext_vector_type(8))) float v8f;

#define N_ROWS   65536      // 64*32*32
#define DIM      256
#define K_CODES  2048
#define TILE_M   16
#define TILE_N   16
#define WAVES_PER_BLOCK 8

// ---------------- workspace layout (bytes) ----------------
// [0,      8K )  float enorm[2048]
// [8K,    16K )  int   counts[2048]
// [16K,  272K )  int   idx[65536]
// [272K, 528K )  float sse_partial[65536]

__global__ void vq_zero_counts(int* __restrict__ counts) {
  int i = blockIdx.x * blockDim.x + threadIdx.x;
  if (i < K_CODES) counts[i] = 0;
}

__global__ void vq_enorm(const float* __restrict__ emb, float* __restrict__ enorm) {
  __shared__ float red[DIM];
  const int k = blockIdx.x;
  const int t = threadIdx.x;
  float v = emb[(size_t)k * DIM + t];
  red[t] = v * v;
  __syncthreads();
  for (int s = DIM / 2; s > 0; s >>= 1) {
    if (t < s) red[t] += red[t + s];
    __syncthreads();
  }
  if (t == 0) enorm[k] = red[0];
}

// One wave handles a 16-row tile of f_emb; sweeps all 2048 codes.
// A-fragment layout (16x4 fp32, MxK): lanes 0-15 and 16-31 both hold M=lane&15;
//   VGPR0/1 hold K = {0,1} (lanes 0-15) or {2,3} (lanes 16-31).
// B-fragment (4x16, KxN): lane holds column N=lane&15, same K split.
// C layout (16x16 fp32): VGPR i, lanes 0-15 -> (M=i, N=lane); lanes 16-31 -> (M=8+i, N=lane-16).
// => each lane sees one code n = 16*kt + (lane&15) across all 8 C slots.
__global__ void __launch_bounds__(256)
vq_argmin(const float* __restrict__ f, const float* __restrict__ emb,
          const float* __restrict__ enorm, int* __restrict__ idx_ws,
          float* __restrict__ idx_out) {
  const int lane = threadIdx.x & 31;
  const int wave = threadIdx.x >> 5;
  const int tile = blockIdx.x * WAVES_PER_BLOCK + wave;
  const int rowBase = tile * TILE_M;
  const int r = lane & 15;
  const int sel = (lane < 16) ? 0 : 2;

  // Register-resident A tile: 16 rows x 256 dims -> 64 v2f fragments (128 VGPRs).
  const float* aP = f + (size_t)(rowBase + r) * DIM + sel;
  v2f a[DIM / 4];
#pragma unroll
  for (int c = 0; c < DIM / 4; ++c)
    a[c] = *(const v2f*)(aP + 4 * c);

  float bestV[8];
  int bestI[8];
#pragma unroll
  for (int i = 0; i < 8; ++i) { bestV[i] = 3.0e38f; bestI[i] = 0; }

  for (int kt = 0; kt < K_CODES / TILE_N; ++kt) {
    const int col = kt * TILE_N + r;                 // code index this lane sees
    const float* bP = emb + (size_t)col * DIM + sel; // streams from L2 (2MB codebook)
    v8f acc = {};
#pragma unroll
    for (int c = 0; c < DIM / 4; ++c) {
      v2f b = *(const v2f*)(bP + 4 * c);
      acc = __builtin_amdgcn_wmma_f32_16x16x4_f32(
          /*neg_a=*/false, a[c], /*neg_b=*/false, b,
          /*c_mod=*/(short)0, acc, /*reuse_a=*/false, /*reuse_b=*/false);
    }
    const float en = enorm[col];
    // argmin proxy: |e|^2 - 2*x.e   (|x|^2 is row-constant, dropped)
#pragma unroll
    for (int i = 0; i < 8; ++i) {
      float d = fmaf(-2.0f, acc[i], en);
      if (d < bestV[i]) { bestV[i] = d; bestI[i] = col; }  // strict < => lowest index wins
    }
  }

  // Cross-lane argmin within each 16-lane half (halves carry disjoint M ranges).
#pragma unroll
  for (int i = 0; i < 8; ++i) {
    float bv = bestV[i];
    int bi = bestI[i];
#pragma unroll
    for (int m = 1; m < 16; m <<= 1) {
      float ov = __shfl_xor(bv, m, 32);
      int oi = __shfl_xor(bi, m, 32);
      if (ov < bv || (ov == bv && oi < bi)) { bv = ov; bi = oi; }
    }
    if (r == 0) {
      const int row = rowBase + i + ((lane < 16) ? 0 : 8);
      idx_ws[row] = bi;
      idx_out[row] = (float)bi;
    }
  }
}

// One block per row: gather code vector, write straight-through forward value,
// deterministic per-row squared error, histogram via int atomics.
__global__ void vq_gather(const float* __restrict__ f, const float* __restrict__ emb,
                          const int* __restrict__ idx_ws, float* __restrict__ qout,
                          int* __restrict__ counts, float* __restrict__ ssep) {
  __shared__ float red[DIM];
  const int row = blockIdx.x;
  const int t = threadIdx.x;
  const int k = idx_ws[row];
  float fv = f[(size_t)row * DIM + t];
  float qv = emb[(size_t)k * DIM + t];
  float diff = qv - fv;
  qout[(size_t)row * DIM + t] = fv + diff;  // f + sg(q - f): forward value
  red[t] = diff * diff;
  __syncthreads();
  for (int s = DIM / 2; s > 0; s >>= 1) {
    if (t < s) red[t] += red[t + s];
    __syncthreads();
  }
  if (t == 0) {
    ssep[row] = red[0];
    atomicAdd(&counts[k], 1);
  }
}

// Deterministic final reductions: SSE over 65536 partials, entropy over 2048 counts.
__global__ void vq_finalize(const float* __restrict__ ssep, const int* __restrict__ counts,
                            float* __restrict__ out_scalars) {
  __shared__ float red[256];
  const int t = threadIdx.x;
  float s = 0.f;
  for (int i = t; i < N_ROWS; i += 256) s += ssep[i];  // fixed order per thread
  red[t] = s;
  __syncthreads();
  for (int st = 128; st > 0; st >>= 1) {
    if (t < st) red[t] += red[t + st];
    __syncthreads();
  }
  const float sse = red[0];
  __syncthreads();

  float h = 0.f;
  for (int k = t; k < K_CODES; k += 256) {
    float p = (float)counts[k] * (1.0f / (float)N_ROWS);
    h += p * logf(p + 1e-10f);
  }
  red[t] = h;
  __syncthreads();
  for (int st = 128; st > 0; st >>= 1) {
    if (t < st) red[t] += red[t + st];
    __syncthreads();
  }
  if (t == 0) {
    float mse = sse / (float)((size_t)N_ROWS * DIM);
    out_scalars[0] = mse * 1.25f;   // q_latent_loss + 0.25 * e_latent_loss (equal forward)
    out_scalars[1] = expf(-red[0]); // perplexity
  }
}

extern "C" void kernel_launch(void* const* d_in, const int* in_sizes, int n_in,
                              void* d_out, int out_size, void* d_ws, size_t ws_size,
                              hipStream_t stream) {
  (void)in_sizes; (void)n_in; (void)out_size; (void)ws_size;
  const float* f_emb = (const float*)d_in[0];    // [65536, 256]
  const float* emb   = (const float*)d_in[1];    // [2048, 256]
  float* out = (float*)d_out;

  char* ws = (char*)d_ws;
  float* enorm  = (float*)(ws);
  int*   counts = (int*)(ws + 8 * 1024);
  int*   idx_ws = (int*)(ws + 16 * 1024);
  float* ssep   = (float*)(ws + 272 * 1024);

  float* qout    = out;                       // [65536*256] quantized_st
  float* scalars = out + (size_t)N_ROWS * DIM; // [2]: latent_loss, perplexity
  float* idx_out = scalars + 2;               // [65536] indices as float

  vq_zero_counts<<<(K_CODES + 255) / 256, 256, 0, stream>>>(counts);
  vq_enorm<<<K_CODES, DIM, 0, stream>>>(emb, enorm);
  vq_argmin<<<N_ROWS / (TILE_M * WAVES_PER_BLOCK), 256, 0, stream>>>(
      f_emb, emb, enorm, idx_ws, idx_out);
  vq_gather<<<N_ROWS, DIM, 0, stream>>>(f_emb, emb, idx_ws, qout, counts, ssep);
  vq_finalize<<<1, 256, 0, stream>>>(ssep, counts, scalars);
}